// Arch9Layer_67053029425455
// MI455X (gfx1250) — compile-verified
//
#include <hip/hip_runtime.h>
#include <hip/hip_bf16.h>

// ---------------------------------------------------------------------------
// MI455X (gfx1250) implementation. wave32, WMMA f32 16x16x4 for all GEMMs.
// Op is memory/atomic-bound (~50 GFLOP vs ~40-80us HBM floor at 23.3 TB/s),
// so we keep full fp32 precision via V_WMMA_F32_16X16X4_F32.
// ---------------------------------------------------------------------------

typedef float v2f __attribute__((ext_vector_type(2)));
typedef float v8f __attribute__((ext_vector_type(8)));

__device__ __forceinline__ v8f wmma_f32_16x16x4(v2f a, v2f b, v8f c) {
  // 8 args: (neg_a, A, neg_b, B, c_mod, C, reuse_a, reuse_b)
  return __builtin_amdgcn_wmma_f32_16x16x4_f32(false, a, false, b, (short)0, c,
                                               false, false);
}

__device__ __forceinline__ void atomic_add_f32(float* p, float v) {
  unsafeAtomicAdd(p, v);  // lowers to global_atomic_add_f32
}

#define HDIM 128
#define EDIM 64
#define LDSP 132  // LDS row pitch (floats): stride 132 -> conflict-free banks

// ---------------------------------------------------------------------------
// Per-edge message + scatter block body. FULL=true: block of 16 edges is
// entirely in range -> no per-element guards, EXEC untouched in hot loop.
// One wave handles 16 edges: D[16,128] = EA[16,64] @ EW[64,128],
// msg = relu(D + eb + x[src]); atomicAdd into agg[dst].
// ---------------------------------------------------------------------------
template <bool FULL>
__device__ __forceinline__ void edge_block(
    const float* __restrict__ x, const int* __restrict__ src,
    const int* __restrict__ dst, const float* __restrict__ ea,
    const float* __restrict__ ew, const float* __restrict__ eb,
    float* __restrict__ agg, int n_edges, int e0, int lane) {
  const int half = lane >> 4, lm = lane & 15;

  // Preload A tile (16 edges x 64): 16 k-steps x v2f = 32 VGPRs
  v2f a[16];
  const int arow = e0 + lm;
  const bool rvalid = FULL || (arow < n_edges);
  const float* ap = ea + (size_t)(rvalid ? arow : 0) * EDIM;
#pragma unroll
  for (int k = 0; k < 16; ++k) {
    const int kk = 4 * k + 2 * half;
    if (FULL) {
      a[k].x = ap[kk];
      a[k].y = ap[kk + 1];
    } else {
      a[k].x = rvalid ? ap[kk] : 0.f;
      a[k].y = rvalid ? ap[kk + 1] : 0.f;
    }
  }

  // Per-lane output edges: vgpr r -> edge e0 + r + 8*half
  int sa[8], da[8];
  bool ev[8];
#pragma unroll
  for (int r = 0; r < 8; ++r) {
    const int e = e0 + r + 8 * half;
    ev[r] = FULL || (e < n_edges);
    const int es = ev[r] ? e : 0;
    sa[r] = src[es];
    da[r] = dst[es];
  }

  for (int nt = 0; nt < 8; ++nt) {
    const int n0 = nt * 16;
    v8f acc = {0.f, 0.f, 0.f, 0.f, 0.f, 0.f, 0.f, 0.f};
#pragma unroll
    for (int k = 0; k < 16; ++k) {
      const int kk = 4 * k + 2 * half;
      v2f b;
      b.x = ew[kk * HDIM + n0 + lm];
      b.y = ew[(kk + 1) * HDIM + n0 + lm];
      acc = wmma_f32_16x16x4(a[k], b, acc);
    }
    const int col = n0 + lm;
    const float bias = eb[col];
#pragma unroll
    for (int r = 0; r < 8; ++r) {
      if (FULL || ev[r]) {
        float v = acc[r] + bias + x[(size_t)sa[r] * HDIM + col];
        v = fmaxf(v, 0.f);
        atomic_add_f32(&agg[(size_t)da[r] * HDIM + col], v);
      }
    }
  }
}

__global__ void edge_msg_scatter(const float* __restrict__ x,
                                 const int* __restrict__ src,
                                 const int* __restrict__ dst,
                                 const float* __restrict__ ea,
                                 const float* __restrict__ ew,
                                 const float* __restrict__ eb,
                                 float* __restrict__ agg, int n_edges) {
  const int e0 = blockIdx.x * 16;
  const int lane = threadIdx.x & 31;
  if (e0 + 16 <= n_edges) {  // wave-uniform branch
    edge_block<true>(x, src, dst, ea, ew, eb, agg, n_edges, e0, lane);
  } else {
    edge_block<false>(x, src, dst, ea, ew, eb, agg, n_edges, e0, lane);
  }
}

// ---------------------------------------------------------------------------
// Kernel: GINE MLP. 16 rows per block (128 thr = 4 waves).
// h = (1+eps)*x + agg ; out = relu(h@W1 + b1)@W2 + b2
// ---------------------------------------------------------------------------
__global__ void gine_mlp(const float* __restrict__ x,
                         const float* __restrict__ agg,
                         const float* __restrict__ epsp,
                         const float* __restrict__ w1,
                         const float* __restrict__ b1,
                         const float* __restrict__ w2,
                         const float* __restrict__ b2,
                         float* __restrict__ out, int n_rows) {
  __shared__ float hbuf[16 * LDSP];
  __shared__ float ybuf[16 * LDSP];
  const int r0 = blockIdx.x * 16;
  const int tid = threadIdx.x;
  const int wave = tid >> 5, lane = tid & 31;
  const int half = lane >> 4, lm = lane & 15;
  const float epsv = 1.0f + epsp[0];

  for (int i = tid; i < 16 * HDIM; i += 128) {
    const int r = i >> 7, c = i & 127;
    const int row = r0 + r;
    float v = 0.f;
    if (row < n_rows)
      v = epsv * x[(size_t)row * HDIM + c] + agg[(size_t)row * HDIM + c];
    hbuf[r * LDSP + c] = v;
  }
  __syncthreads();

  // GEMM1 -> relu -> ybuf
  for (int t = 0; t < 2; ++t) {
    const int n0 = (wave * 2 + t) * 16;
    v8f acc = {0.f, 0.f, 0.f, 0.f, 0.f, 0.f, 0.f, 0.f};
#pragma unroll
    for (int k = 0; k < 32; ++k) {
      const int kk = 4 * k + 2 * half;
      v2f a, b;
      a.x = hbuf[lm * LDSP + kk];
      a.y = hbuf[lm * LDSP + kk + 1];
      b.x = w1[kk * HDIM + n0 + lm];
      b.y = w1[(kk + 1) * HDIM + n0 + lm];
      acc = wmma_f32_16x16x4(a, b, acc);
    }
    const int col = n0 + lm;
    const float bias = b1[col];
#pragma unroll
    for (int r = 0; r < 8; ++r)
      ybuf[(r + 8 * half) * LDSP + col] = fmaxf(acc[r] + bias, 0.f);
  }
  __syncthreads();

  // GEMM2 -> out
  for (int t = 0; t < 2; ++t) {
    const int n0 = (wave * 2 + t) * 16;
    v8f acc = {0.f, 0.f, 0.f, 0.f, 0.f, 0.f, 0.f, 0.f};
#pragma unroll
    for (int k = 0; k < 32; ++k) {
      const int kk = 4 * k + 2 * half;
      v2f a, b;
      a.x = ybuf[lm * LDSP + kk];
      a.y = ybuf[lm * LDSP + kk + 1];
      b.x = w2[kk * HDIM + n0 + lm];
      b.y = w2[(kk + 1) * HDIM + n0 + lm];
      acc = wmma_f32_16x16x4(a, b, acc);
    }
    const int col = n0 + lm;
    const float bias = b2[col];
#pragma unroll
    for (int r = 0; r < 8; ++r) {
      const int row = r0 + r + 8 * half;
      if (row < n_rows) out[(size_t)row * HDIM + col] = acc[r] + bias;
    }
  }
}

// ---------------------------------------------------------------------------
// Kernel: non-root projection: out = x@Ws + bs + x_root@Wr + br
// ---------------------------------------------------------------------------
__global__ void dual_proj(const float* __restrict__ x,
                          const int* __restrict__ sub_batch,
                          const int* __restrict__ kptr,
                          const float* __restrict__ wsf,
                          const float* __restrict__ bsf,
                          const float* __restrict__ wrt,
                          const float* __restrict__ brt,
                          float* __restrict__ out, int n_rows) {
  __shared__ float abuf[16 * LDSP];
  __shared__ float rbuf[16 * LDSP];
  const int r0 = blockIdx.x * 16;
  const int tid = threadIdx.x;
  const int wave = tid >> 5, lane = tid & 31;
  const int half = lane >> 4, lm = lane & 15;
  const int K = kptr[0];

  for (int i = tid; i < 16 * HDIM; i += 128) {
    const int r = i >> 7, c = i & 127;
    const int row = r0 + r;
    float av = 0.f, rv = 0.f;
    if (row < n_rows) {
      av = x[(size_t)row * HDIM + c];
      const int rootrow = sub_batch[row] * K;
      rv = x[(size_t)rootrow * HDIM + c];
    }
    abuf[r * LDSP + c] = av;
    rbuf[r * LDSP + c] = rv;
  }
  __syncthreads();

  for (int t = 0; t < 2; ++t) {
    const int n0 = (wave * 2 + t) * 16;
    v8f acc = {0.f, 0.f, 0.f, 0.f, 0.f, 0.f, 0.f, 0.f};
#pragma unroll
    for (int k = 0; k < 32; ++k) {
      const int kk = 4 * k + 2 * half;
      v2f a, b;
      a.x = abuf[lm * LDSP + kk];
      a.y = abuf[lm * LDSP + kk + 1];
      b.x = wsf[kk * HDIM + n0 + lm];
      b.y = wsf[(kk + 1) * HDIM + n0 + lm];
      acc = wmma_f32_16x16x4(a, b, acc);
    }
#pragma unroll
    for (int k = 0; k < 32; ++k) {
      const int kk = 4 * k + 2 * half;
      v2f a, b;
      a.x = rbuf[lm * LDSP + kk];
      a.y = rbuf[lm * LDSP + kk + 1];
      b.x = wrt[kk * HDIM + n0 + lm];
      b.y = wrt[(kk + 1) * HDIM + n0 + lm];
      acc = wmma_f32_16x16x4(a, b, acc);
    }
    const int col = n0 + lm;
    const float bias = bsf[col] + brt[col];
#pragma unroll
    for (int r = 0; r < 8; ++r) {
      const int row = r0 + r + 8 * half;
      if (row < n_rows) out[(size_t)row * HDIM + col] = acc[r] + bias;
    }
  }
}

// ---------------------------------------------------------------------------
// BN statistics: per-column sum / sum-of-squares (grid-strided, coalesced)
// ---------------------------------------------------------------------------
__global__ void bn_stats(const float* __restrict__ v, int n_rows,
                         float* __restrict__ stats) {
  const int col = threadIdx.x;  // 0..127
  float s = 0.f, s2 = 0.f;
  for (int row = blockIdx.x; row < n_rows; row += gridDim.x) {
    const float x = v[(size_t)row * HDIM + col];
    s += x;
    s2 += x * x;
  }
  atomic_add_f32(&stats[col], s);
  atomic_add_f32(&stats[HDIM + col], s2);
}

__global__ void bn_finalize(float* __restrict__ stats, int n_rows) {
  const int c = threadIdx.x;
  const float inv_n = 1.0f / (float)n_rows;
  const float mu = stats[c] * inv_n;
  const float var = stats[HDIM + c] * inv_n - mu * mu;
  stats[c] = mu;
  stats[HDIM + c] = rsqrtf(var + 1e-5f);
}

// ---------------------------------------------------------------------------
// Root scatter-mean onto canonical ids
// ---------------------------------------------------------------------------
__global__ void root_scatter(const float* __restrict__ x,
                             const int* __restrict__ root_idx,
                             const int* __restrict__ node_ids,
                             float* __restrict__ ssum, float* __restrict__ cnt,
                             int S) {
  const int i = blockIdx.x * blockDim.x + threadIdx.x;
  if (i >= S * HDIM) return;
  const int r = i >> 7, c = i & 127;
  const int fi = root_idx[r];
  const int rid = node_ids[fi];
  atomic_add_f32(&ssum[(size_t)rid * HDIM + c], x[(size_t)fi * HDIM + c]);
  if (c == 0) atomic_add_f32(&cnt[rid], 1.0f);
}

__global__ void canon_finalize(const float* __restrict__ ssum,
                               const float* __restrict__ cnt,
                               float* __restrict__ hc, int NT) {
  const int i = blockIdx.x * blockDim.x + threadIdx.x;
  if (i >= NT * HDIM) return;
  const float c = cnt[i >> 7];
  hc[i] = (c > 0.f) ? ssum[i] / fmaxf(c, 1.0f) : 0.f;
}

// ---------------------------------------------------------------------------
// Final fuse: BN-apply x2, root/non-root select, relu, valid mask
// ---------------------------------------------------------------------------
__global__ void fuse_out(const float* __restrict__ h1pre,
                         const float* __restrict__ bn1s,
                         const float* __restrict__ bn1g,
                         const float* __restrict__ bn1b,
                         const float* __restrict__ hinterpre,
                         const float* __restrict__ bn2s,
                         const float* __restrict__ bn2g,
                         const float* __restrict__ bn2b,
                         const float* __restrict__ hnr,
                         const unsigned char* __restrict__ valid,
                         const int* __restrict__ node_ids,
                         const int* __restrict__ kptr,
                         float* __restrict__ out, int n_flat) {
  const int i = blockIdx.x * blockDim.x + threadIdx.x;
  if (i >= n_flat * HDIM) return;
  const int row = i >> 7, c = i & 127;
  const float vmask = valid[row] ? 1.f : 0.f;  // np.bool_ = 1 byte
  const float h1 = (h1pre[i] - bn1s[c]) * bn1s[HDIM + c] * bn1g[c] + bn1b[c];
  const int K = kptr[0];
  float other;
  if ((row % K) == 0) {
    int cid = node_ids[row];
    if (cid < 0) cid = 0;
    const float hp = hinterpre[(size_t)cid * HDIM + c];
    other = (hp - bn2s[c]) * bn2s[HDIM + c] * bn2g[c] + bn2b[c];
  } else {
    other = hnr[i];
  }
  out[i] = fmaxf(h1 + other, 0.f) * vmask;
}

// ---------------------------------------------------------------------------
extern "C" void kernel_launch(void* const* d_in, const int* in_sizes, int n_in,
                              void* d_out, int out_size, void* d_ws,
                              size_t ws_size, hipStream_t stream) {
  const int NT = 50000;  // canonical node count (fixed by problem setup)
  const int n_flat = in_sizes[0] / HDIM;
  const int EI = in_sizes[1] / 2;
  const int EE = in_sizes[6] / 2;
  const int S = in_sizes[11];

  const float* h_flat = (const float*)d_in[0];
  const int* intra_ei = (const int*)d_in[1];
  const float* ea_flat = (const float*)d_in[2];
  const unsigned char* valid = (const unsigned char*)d_in[3];
  const int* node_ids = (const int*)d_in[4];
  const int* edge_index = (const int*)d_in[6];
  const float* edge_attr = (const float*)d_in[7];
  const int* sub_batch = (const int*)d_in[8];
  const int* kptr = (const int*)d_in[10];
  const int* root_idx = (const int*)d_in[11];
  const float* eps_intra = (const float*)d_in[12];
  const float* ew_intra = (const float*)d_in[13];
  const float* eb_intra = (const float*)d_in[14];
  const float* w1_intra = (const float*)d_in[15];
  const float* b1_intra = (const float*)d_in[16];
  const float* w2_intra = (const float*)d_in[17];
  const float* b2_intra = (const float*)d_in[18];
  const float* bn1_g = (const float*)d_in[19];
  const float* bn1_b = (const float*)d_in[20];
  const float* self_w = (const float*)d_in[21];
  const float* self_b = (const float*)d_in[22];
  const float* root_w = (const float*)d_in[23];
  const float* root_b = (const float*)d_in[24];
  const float* eps_inter = (const float*)d_in[25];
  const float* ew_inter = (const float*)d_in[26];
  const float* eb_inter = (const float*)d_in[27];
  const float* w1_inter = (const float*)d_in[28];
  const float* b1_inter = (const float*)d_in[29];
  const float* w2_inter = (const float*)d_in[30];
  const float* b2_inter = (const float*)d_in[31];
  const float* bn2_g = (const float*)d_in[32];
  const float* bn2_b = (const float*)d_in[33];

  // Workspace layout (floats), with aliasing to keep footprint ~240MB.
  float* W = (float*)d_ws;
  size_t o = 0;
  float* agg_intra = W + o;  // later reused as h_nonroot
  o += (size_t)n_flat * HDIM;
  float* h1_pre = W + o;
  o += (size_t)n_flat * HDIM;
  float* ssum = W + o;  // later reused as agg_inter
  o += (size_t)NT * HDIM;
  float* cnt = W + o;
  o += NT;
  float* h_canon = W + o;
  o += (size_t)NT * HDIM;
  float* h_inter = W + o;
  o += (size_t)NT * HDIM;
  float* bn1s = W + o;
  o += 2 * HDIM;
  float* bn2s = W + o;
  o += 2 * HDIM;
  (void)o;

  // Zero accumulators (ssum+cnt contiguous; bn1s+bn2s contiguous)
  hipMemsetAsync(agg_intra, 0, (size_t)n_flat * HDIM * sizeof(float), stream);
  hipMemsetAsync(ssum, 0, ((size_t)NT * HDIM + NT) * sizeof(float), stream);
  hipMemsetAsync(bn1s, 0, 4 * HDIM * sizeof(float), stream);

  // (1) intra GINE: edge messages + scatter, then node MLP
  edge_msg_scatter<<<(EI + 15) / 16, 32, 0, stream>>>(
      h_flat, intra_ei, intra_ei + EI, ea_flat, ew_intra, eb_intra, agg_intra,
      EI);
  gine_mlp<<<(n_flat + 15) / 16, 128, 0, stream>>>(
      h_flat, agg_intra, eps_intra, w1_intra, b1_intra, w2_intra, b2_intra,
      h1_pre, n_flat);
  bn_stats<<<512, HDIM, 0, stream>>>(h1_pre, n_flat, bn1s);
  bn_finalize<<<1, HDIM, 0, stream>>>(bn1s, n_flat);

  // scatter-mean roots -> canonical
  root_scatter<<<(S * HDIM + 255) / 256, 256, 0, stream>>>(
      h_flat, root_idx, node_ids, ssum, cnt, S);
  canon_finalize<<<(NT * HDIM + 255) / 256, 256, 0, stream>>>(ssum, cnt,
                                                              h_canon, NT);

  // (2) inter GINE on canonical graph (reuse ssum as agg_inter)
  hipMemsetAsync(ssum, 0, (size_t)NT * HDIM * sizeof(float), stream);
  edge_msg_scatter<<<(EE + 15) / 16, 32, 0, stream>>>(
      h_canon, edge_index, edge_index + EE, edge_attr, ew_inter, eb_inter,
      ssum, EE);
  gine_mlp<<<(NT + 15) / 16, 128, 0, stream>>>(h_canon, ssum, eps_inter,
                                               w1_inter, b1_inter, w2_inter,
                                               b2_inter, h_inter, NT);
  bn_stats<<<512, HDIM, 0, stream>>>(h_inter, NT, bn2s);
  bn_finalize<<<1, HDIM, 0, stream>>>(bn2s, NT);

  // non-root branch (reuse agg_intra as h_nonroot)
  dual_proj<<<(n_flat + 15) / 16, 128, 0, stream>>>(
      h_flat, sub_batch, kptr, self_w, self_b, root_w, root_b, agg_intra,
      n_flat);

  // final combine
  fuse_out<<<((size_t)n_flat * HDIM + 255) / 256, 256, 0, stream>>>(
      h1_pre, bn1s, bn1_g, bn1_b, h_inter, bn2s, bn2_g, bn2_b, agg_intra,
      valid, node_ids, kptr, (float*)d_out, n_flat);
}